// SoftmaxTreePrediction_86122684219469
// MI455X (gfx1250) — compile-verified
//
#include <hip/hip_runtime.h>
#include <stdint.h>

// Problem constants (match reference setup_inputs)
#define NSAMP 16
#define HW    4096      // 64*64
#define CCH   2320      // node_count
#define COUT  2321      // + append_max channel
#define NGRP  289       // number of groups (1 root + 32 + 256)
#define NNODE 2320
#define SMT_THRESH 0.5f

// Native clang vector type: accepted by __builtin_nontemporal_store
typedef float __attribute__((ext_vector_type(4))) v4f;

// ---------------------------------------------------------------------------
// Kernel 1: zero-fill the 608MB output with non-temporal 128-bit stores
// (output is far larger than the 192MB L2 -> keep it out of the cache).
// ---------------------------------------------------------------------------
__global__ __launch_bounds__(256) void smt_fill_zero(v4f* __restrict__ out4,
                                                     size_t n4) {
    size_t i      = (size_t)blockIdx.x * blockDim.x + threadIdx.x;
    size_t stride = (size_t)gridDim.x * blockDim.x;
    v4f z = {0.f, 0.f, 0.f, 0.f};
    for (; i < n4; i += stride)
        __builtin_nontemporal_store(z, out4 + i);
}

// ---------------------------------------------------------------------------
// Kernel 2: per-lane DFS over the softmax tree.
// Tree metadata (~21KB) is staged into LDS with gfx1250 async global->LDS
// copies (ASYNCcnt), then each thread (one per (n,h,w) lane) traverses.
// ---------------------------------------------------------------------------
__global__ __launch_bounds__(256) void smt_traverse(
    const float* __restrict__ conf, const float* __restrict__ obj,
    const int* __restrict__ go, const int* __restrict__ gs,
    const int* __restrict__ child, const int* __restrict__ cs,
    float* __restrict__ out) {
    __shared__ int s_go[NGRP];
    __shared__ int s_gs[NGRP];
    __shared__ int s_child[NNODE];
    __shared__ int s_cs[NNODE];

    const int tid = threadIdx.x;

    // Async copy metadata into LDS (CDNA5: global_load_async_to_lds_*,
    // tracked with ASYNCcnt, drained with s_wait_asynccnt).
    for (int i = tid; i < NGRP; i += 256) {
        uint32_t l0 = (uint32_t)(uintptr_t)&s_go[i];
        uint64_t g0 = (uint64_t)(uintptr_t)(go + i);
        asm volatile("global_load_async_to_lds_b32 %0, %1, off"
                     :: "v"(l0), "v"(g0) : "memory");
        uint32_t l1 = (uint32_t)(uintptr_t)&s_gs[i];
        uint64_t g1 = (uint64_t)(uintptr_t)(gs + i);
        asm volatile("global_load_async_to_lds_b32 %0, %1, off"
                     :: "v"(l1), "v"(g1) : "memory");
    }
    for (int i = tid; i < NNODE; i += 256) {
        uint32_t l0 = (uint32_t)(uintptr_t)&s_child[i];
        uint64_t g0 = (uint64_t)(uintptr_t)(child + i);
        asm volatile("global_load_async_to_lds_b32 %0, %1, off"
                     :: "v"(l0), "v"(g0) : "memory");
        uint32_t l1 = (uint32_t)(uintptr_t)&s_cs[i];
        uint64_t g1 = (uint64_t)(uintptr_t)(cs + i);
        asm volatile("global_load_async_to_lds_b32 %0, %1, off"
                     :: "v"(l1), "v"(g1) : "memory");
    }
    asm volatile("s_wait_asynccnt 0" ::: "memory");
    __syncthreads();

    const int id = blockIdx.x * 256 + tid;   // 0 .. N*HW-1
    const int n  = id >> 12;                 // id / 4096
    const int hw = id & (HW - 1);

    const float o = __builtin_nontemporal_load(obj + id);
    const float* cbase = conf + (size_t)n * CCH * HW + hw;
    float*       obase = out  + (size_t)n * COUT * HW + hw;

    // DFS stack (max depth: root pushes 2, each level-2 node pushes 1)
    int   stk_g[8];
    float stk_p[8];
    int   stk_a[8];
    int ptr = 1;
    stk_g[0] = 0;            // root group
    stk_p[0] = o;            // path prob starts at objectness
    stk_a[0] = -1;           // no parent argmax yet
    float topmax = 0.f;

    while (ptr > 0) {
        --ptr;
        const int   g   = stk_g[ptr];
        const float pp  = stk_p[ptr];
        const int   pa  = stk_a[ptr];
        const int   off = s_go[g];
        const int   sz  = s_gs[g];

        // argmax within the group (first max wins, matching jnp.argmax)
        float best = -__builtin_inff();
        int   bj   = 0;
        for (int k = 0; k < sz; ++k) {
            float v = __builtin_nontemporal_load(cbase + (size_t)(off + k) * HW);
            if (v > best) { best = v; bj = k; }
        }
        const float p    = pp * best;
        const int   amax = off + bj;

        if (p > SMT_THRESH) {
            const int c = s_child[amax];
            if (c < 0) {
                // leaf hit: write cumulative path probability at argmax node
                obase[(size_t)amax * HW] = p;
                topmax = fmaxf(topmax, p);
            } else {
                // descend: push subgroups c .. c+child_sizes
                const int npush = s_cs[amax] + 1;
                for (int t = 0; t < npush; ++t) {
                    stk_g[ptr] = c + t;
                    stk_p[ptr] = p;
                    stk_a[ptr] = amax;
                    ++ptr;
                }
            }
        } else if (pa >= 0) {
            // pruned: write parent path probability at parent argmax node
            obase[(size_t)pa * HW] = pp;
            topmax = fmaxf(topmax, pp);
        }
    }
    // append_max channel
    obase[(size_t)CCH * HW] = topmax;
}

// ---------------------------------------------------------------------------
extern "C" void kernel_launch(void* const* d_in, const int* in_sizes, int n_in,
                              void* d_out, int out_size, void* d_ws, size_t ws_size,
                              hipStream_t stream) {
    (void)in_sizes; (void)n_in; (void)d_ws; (void)ws_size;

    const float* conf  = (const float*)d_in[0];
    const float* obj   = (const float*)d_in[1];
    const int*   go    = (const int*)d_in[2];
    const int*   gs    = (const int*)d_in[3];
    const int*   child = (const int*)d_in[4];
    const int*   cs    = (const int*)d_in[5];
    float*       out   = (float*)d_out;

    // out_size = 16*2321*4096, divisible by 4 -> v4f fill
    size_t n4 = (size_t)out_size / 4;
    smt_fill_zero<<<4096, 256, 0, stream>>>((v4f*)out, n4);

    smt_traverse<<<(NSAMP * HW) / 256, 256, 0, stream>>>(
        conf, obj, go, gs, child, cs, out);
}